// ACTLossHead_63041529971094
// MI455X (gfx1250) — compile-verified
//
#include <hip/hip_runtime.h>
#include <hip/hip_bf16.h>
#include <math.h>

#define BB 128
#define SS 4096
#define VV 64

typedef __attribute__((ext_vector_type(2))) float v2f;
typedef __attribute__((ext_vector_type(8))) float v8f;

// ---------------------------------------------------------------------------
// Kernel Z: zero per-row nll accumulators (graph-replay safe init)
// ---------------------------------------------------------------------------
__global__ void zero_kernel(float* rowNll) {
    rowNll[threadIdx.x] = 0.0f;
}

// ---------------------------------------------------------------------------
// Kernel A: stream logits once. One token per lane (V=64 -> 16 float4 loads,
// all lane-local: max, argmax(first), logsumexp, nll). Block-reduce nll,
// one atomicAdd per block into rowNll[b]. Store pred id (u8).
// ---------------------------------------------------------------------------
__global__ __launch_bounds__(256) void ce_kernel(const float* __restrict__ logits,
                                                 const int* __restrict__ labels,
                                                 unsigned char* __restrict__ preds,
                                                 float* __restrict__ rowNll) {
    const int b = blockIdx.y;
    const int s = blockIdx.x * 256 + threadIdx.x;     // gridDim.x = SS/256
    const size_t t = (size_t)b * SS + s;
    const float4* x4 = (const float4*)(logits + t * VV);

    float4 v[16];
#pragma unroll
    for (int i = 0; i < 16; ++i) v[i] = x4[i];

    // argmax (first occurrence) + max
    float m = v[0].x; int am = 0;
#pragma unroll
    for (int i = 0; i < 16; ++i) {
        if (v[i].x > m) { m = v[i].x; am = 4 * i + 0; }
        if (v[i].y > m) { m = v[i].y; am = 4 * i + 1; }
        if (v[i].z > m) { m = v[i].z; am = 4 * i + 2; }
        if (v[i].w > m) { m = v[i].w; am = 4 * i + 3; }
    }

    float se = 0.0f;
#pragma unroll
    for (int i = 0; i < 16; ++i) {
        se += expf(v[i].x - m);
        se += expf(v[i].y - m);
        se += expf(v[i].z - m);
        se += expf(v[i].w - m);
    }
    const float lse = m + logf(se);

    const int lab = labels[t];
    float nll = 0.0f;
    if (lab != 0) {
        const float xl = (logits + t * VV)[lab];      // L0-hit reload, avoids VGPR spill
        nll = lse - xl;
    }
    preds[t] = (unsigned char)am;

    // block reduction of nll
    __shared__ float sred[256];
    sred[threadIdx.x] = nll;
    __syncthreads();
    for (int o = 128; o > 0; o >>= 1) {
        if (threadIdx.x < (unsigned)o) sred[threadIdx.x] += sred[threadIdx.x + o];
        __syncthreads();
    }
    if (threadIdx.x == 0) atomicAdd(rowNll + b, sred[0]);
}

// ---------------------------------------------------------------------------
// Kernel B: one block per row. Marker scan + set-membership + per-row stats.
// Emits rowq[b][0..15] (quantities to be summed over rows by WMMA kernel).
// ---------------------------------------------------------------------------
__global__ __launch_bounds__(256) void row_kernel(const int* __restrict__ labels,
                                                  const unsigned char* __restrict__ preds,
                                                  const float* __restrict__ rowNll,
                                                  const float* __restrict__ qhalt,
                                                  const unsigned char* __restrict__ halted,
                                                  const int* __restrict__ steps,
                                                  float* __restrict__ rowq) {
    const int b = blockIdx.x;
    const int tid = threadIdx.x;
    const int* lab = labels + (size_t)b * SS;
    const unsigned char* prd = preds + (size_t)b * SS;

    __shared__ int sbuf[256];

    auto redMin = [&](int v) -> int {
        sbuf[tid] = v; __syncthreads();
        for (int o = 128; o > 0; o >>= 1) {
            if (tid < o) sbuf[tid] = min(sbuf[tid], sbuf[tid + o]);
            __syncthreads();
        }
        int r = sbuf[0]; __syncthreads(); return r;
    };
    auto redSum = [&](int v) -> int {
        sbuf[tid] = v; __syncthreads();
        for (int o = 128; o > 0; o >>= 1) {
            if (tid < o) sbuf[tid] += sbuf[tid + o];
            __syncthreads();
        }
        int r = sbuf[0]; __syncthreads(); return r;
    };
    auto redOr = [&](int v) -> int {
        sbuf[tid] = v; __syncthreads();
        for (int o = 128; o > 0; o >>= 1) {
            if (tid < o) sbuf[tid] |= sbuf[tid + o];
            __syncthreads();
        }
        int r = sbuf[0]; __syncthreads(); return r;
    };

    // pass 1: markers + counts
    int f28 = SS, f30 = SS, cnt = 0, corr = 0;
#pragma unroll
    for (int i = 0; i < SS / 256; ++i) {
        const int s = tid + i * 256;
        const int l = lab[s];
        const int p = prd[s];
        if (l == 28) f28 = min(f28, s);
        if (l == 30) f30 = min(f30, s);
        if (l != 0) { cnt++; if (p == l) corr++; }
    }
    const int F28 = redMin(f28);
    const int F30 = redMin(f30);
    const int CNT = redSum(cnt);
    const int CORR = redSum(corr);

    const bool has_f = F28 < SS;
    const int f_start = (has_f ? F28 : 0) + 1;       // argmax(is28)+1 (0 if none)
    const bool has_t = F30 < SS;
    const int t_idx = has_t ? F30 : 0;

    // pass 2a: first valid delimiter strictly after f_start
    int fe = SS;
#pragma unroll
    for (int i = 0; i < SS / 256; ++i) {
        const int s = tid + i * 256;
        const int l = lab[s];
        if ((l == 31 || l == 29) && s > f_start) fe = min(fe, s);
    }
    const int FE = redMin(fe);
    const bool has_d = FE < SS;
    const int f_end = has_d ? FE : 0;

    // pass 2b: set-membership bit masks (token ids 2..27 -> bits 0..25)
    unsigned lset = 0, pset = 0, fset = 0;
#pragma unroll
    for (int i = 0; i < SS / 256; ++i) {
        const int s = tid + i * 256;
        const int l = lab[s];
        const int p = prd[s];
        const bool slice = has_t ? (s > t_idx) : true;
        if (slice && l >= 2 && l <= 27) lset |= 1u << (l - 2);
        if (slice && p >= 2 && p <= 27) pset |= 1u << (p - 2);
        const bool fm = has_f && has_d && (s >= f_start) && (s < f_end);
        if (fm && l >= 2 && l <= 27) fset |= 1u << (l - 2);
    }
    const unsigned LSET = (unsigned)redOr((int)lset);
    const unsigned PSET = (unsigned)redOr((int)pset);
    const unsigned FSET = (unsigned)redOr((int)fset);

    if (tid == 0) {
        const int div = max(CNT, 1);
        const bool seqcorr = (CORR == CNT);
        const unsigned missing = LSET & ~PSET;
        const unsigned halluc  = PSET & ~LSET & ~FSET;
        const bool set_ok = (missing == 0u) && (halluc == 0u);

        const float q = qhalt[b];
        const bool hal = halted[b] != 0;
        const bool valid = hal && (CNT > 0);

        // softplus(q) - q*y  (overflow-safe)
        const float y = seqcorr ? 1.0f : 0.0f;
        const float sp = fmaxf(q, 0.0f) + log1pf(expf(-fabsf(q)));
        const float qh_c = sp - q * y;

        float* r = rowq + b * 16;
        r[0] = rowNll[b] / (float)div;                               // lm_loss
        r[1] = qh_c;                                                 // q_halt_loss
        r[2] = valid ? 1.0f : 0.0f;                                  // m_count
        r[3] = valid ? (float)CORR / (float)div : 0.0f;              // m_acc
        r[4] = (valid && seqcorr) ? 1.0f : 0.0f;                     // m_exact
        r[5] = (valid && set_ok) ? 1.0f : 0.0f;                      // m_set
        r[6] = (valid && ((q >= 0.0f) == seqcorr)) ? 1.0f : 0.0f;    // m_qhalt
        r[7] = valid ? (float)steps[b] : 0.0f;                       // m_steps
        r[8] = 0.0f; r[9] = 0.0f; r[10] = 0.0f; r[11] = 0.0f;
        r[12] = 0.0f; r[13] = 0.0f; r[14] = 0.0f; r[15] = 0.0f;
    }
}

// ---------------------------------------------------------------------------
// Kernel C: single wave32. Column-sum of rowq[128][16] on the matrix engine:
// 32 accumulating V_WMMA_F32_16X16X4_F32 with A = ones(16x4), B = 4 rows x 16
// quantities per step. D[m][n] = sum over all 128 rows of quantity n.
// B layout (4x16 f32): VGPR0 = rows {0 (lanes0-15), 2 (lanes16-31)},
//                      VGPR1 = rows {1, 3}. EXEC stays all-ones (cndmask only).
// ---------------------------------------------------------------------------
__global__ __launch_bounds__(32) void final_kernel(const float* __restrict__ rowq,
                                                   float* __restrict__ out) {
    const int lane = threadIdx.x;                 // 0..31, full wave
    const int n = lane & 15;
    const int roff = (lane < 16) ? 0 : 2;

    v2f a; a.x = 1.0f; a.y = 1.0f;                // ones A-matrix (16x4)
    v8f c = {};

    for (int g = 0; g < 32; ++g) {
        const int r0 = 4 * g + roff;
        v2f bvec;
        bvec.x = rowq[r0 * 16 + n];               // B rows 0/2
        bvec.y = rowq[(r0 + 1) * 16 + n];         // B rows 1/3
        c = __builtin_amdgcn_wmma_f32_16x16x4_f32(
                /*neg_a=*/false, a, /*neg_b=*/false, bvec,
                /*c_mod=*/(short)0, c, /*reuse_a=*/false, /*reuse_b=*/false);
    }

    // D VGPR0, lanes 0-15: M=0, N=lane  -> lane n holds total of quantity n
    __shared__ float q[16];
    if (lane < 16) q[lane] = c[0];
    __syncthreads();
    if (lane == 0) {
        const float lm = q[0], qh = q[1];
        out[0] = lm + 0.5f * qh;   // total_loss
        out[1] = lm;               // lm_loss
        out[2] = qh;               // q_halt_loss
        out[3] = q[2];             // m_count
        out[4] = q[3];             // m_acc
        out[5] = q[4];             // m_exact
        out[6] = q[5];             // m_set
        out[7] = q[6];             // m_qhalt
        out[8] = q[7];             // m_steps
    }
}

// ---------------------------------------------------------------------------
extern "C" void kernel_launch(void* const* d_in, const int* in_sizes, int n_in,
                              void* d_out, int out_size, void* d_ws, size_t ws_size,
                              hipStream_t stream) {
    const float* logits        = (const float*)d_in[0];
    const int* labels          = (const int*)d_in[1];
    const float* q_halt        = (const float*)d_in[2];
    const unsigned char* halted= (const unsigned char*)d_in[3];   // jnp.bool_ = 1 byte
    const int* steps           = (const int*)d_in[4];
    float* out                 = (float*)d_out;

    char* ws = (char*)d_ws;
    float* rowNll        = (float*)ws;                       // 128 floats
    float* rowq          = (float*)(ws + 512);               // 128*16 floats
    unsigned char* preds = (unsigned char*)(ws + 512 + 8192);// B*S bytes

    zero_kernel<<<1, BB, 0, stream>>>(rowNll);

    dim3 gA(SS / 256, BB);
    ce_kernel<<<gA, 256, 0, stream>>>(logits, labels, preds, rowNll);

    row_kernel<<<BB, 256, 0, stream>>>(labels, preds, rowNll, q_halt, halted, steps, rowq);

    final_kernel<<<1, 32, 0, stream>>>(rowq, out);
}